// TripletLoss_fix2_9955734192553
// MI455X (gfx1250) — compile-verified
//
#include <hip/hip_runtime.h>
#include <hip/hip_bf16.h>
#include <math.h>

typedef __attribute__((ext_vector_type(2))) float v2f;
typedef __attribute__((ext_vector_type(8))) float v8f;

#define B_ROWS        16384
#define D_COLS        1024
#define ROWS_PER_TILE 16
#define KSPLIT        8                          // K-slices per row tile
#define KSLICE        (D_COLS / KSPLIT)          // 128 columns per wave
#define WAVES_PER_BLOCK 8
#define N_TILES       (B_ROWS / ROWS_PER_TILE)   // 1024
#define N_WAVES       (N_TILES * KSPLIT)         // 8192
#define STAGE2_BLOCKS (B_ROWS / 256)             // 64

#define U        4                               // WMMA chunks per pipeline stage
#define STAGE_K  (U * 4)                         // 16 K-values per stage
#define NSTAGES  (KSLICE / STAGE_K)              // 8 stages

// ---------------------------------------------------------------------------
// Stage 1: wave w handles row-tile (w>>3), K-slice (w&7). Software-pipelined:
// loads for stage s+1 are issued before the WMMA work of stage s, keeping
// ~12 b64 loads in flight per wave. Row sums accumulate via
// V_WMMA_F32_16X16X4_F32 with an all-ones B matrix (free row reduction).
// ---------------------------------------------------------------------------
__global__ __launch_bounds__(256) void triplet_stage1(
    const float* __restrict__ a, const float* __restrict__ p,
    const float* __restrict__ n, float* __restrict__ partial)
{
    const int lane = threadIdx.x & 31;
    const int wave = blockIdx.x * WAVES_PER_BLOCK + (threadIdx.x >> 5);
    const int tile = wave >> 3;                  // row tile index
    const int ks   = wave & (KSPLIT - 1);        // K-slice index

    const int m  = lane & 15;                    // M index within 16-row tile
    const int hi = lane >> 4;                    // lanes 16-31 carry K=2,3 per chunk
    const long rowBase = (long)(tile * ROWS_PER_TILE + m) * (long)D_COLS;
    const int kBeg = ks * KSLICE;
    const int kOff = hi * 2;

    v8f cp = {0.f,0.f,0.f,0.f,0.f,0.f,0.f,0.f};
    v8f cn = {0.f,0.f,0.f,0.f,0.f,0.f,0.f,0.f};
    const v2f ones = {1.0f, 1.0f};

    v2f A[2][U], P[2][U], N[2][U];               // double-buffered load regs

    auto loadStage = [&](int buf, int k0) {
        #pragma unroll
        for (int j = 0; j < U; ++j) {
            const long idx = rowBase + (long)(k0 + 4 * j + kOff);
            A[buf][j] = *(const v2f*)(a + idx);  // global_load_b64
            P[buf][j] = *(const v2f*)(p + idx);
            N[buf][j] = *(const v2f*)(n + idx);
        }
    };
    auto compStage = [&](int buf) {
        #pragma unroll
        for (int j = 0; j < U; ++j) {
            v2f dp = A[buf][j] - P[buf][j];
            v2f dn = A[buf][j] - N[buf][j];
            v2f dp2 = dp * dp;
            v2f dn2 = dn * dn;
            cp = __builtin_amdgcn_wmma_f32_16x16x4_f32(
                    false, dp2, false, ones, (short)0, cp, false, false);
            cn = __builtin_amdgcn_wmma_f32_16x16x4_f32(
                    false, dn2, false, ones, (short)0, cn, false, false);
        }
    };

    loadStage(0, kBeg);                          // prologue
    #pragma unroll
    for (int s = 0; s < NSTAGES; ++s) {
        if (s + 1 < NSTAGES)
            loadStage((s + 1) & 1, kBeg + (s + 1) * STAGE_K);  // prefetch next
        compStage(s & 1);                                      // compute current
    }

    // C layout: VGPR r = rowsum for M=r (lanes 0-15) / M=r+8 (lanes 16-31).
    // Per-wave scratch block of 32 floats:
    //   [0..15]  = dp^2 partials rows 0..15
    //   [16..31] = dn^2 partials rows 0..15
    if (lane == 0 || lane == 16) {
        float* base = partial + ((long)wave << 5) + hi * 8;
        #pragma unroll
        for (int r = 0; r < 8; ++r) {
            base[r]      = cp[r];
            base[16 + r] = cn[r];
        }
    }
}

// ---------------------------------------------------------------------------
// Stage 2: one thread per row. Combine the KSPLIT partial squared sums,
// sqrt, stable softplus, then fixed-order block reduction.
// ---------------------------------------------------------------------------
__global__ __launch_bounds__(256) void triplet_stage2(
    const float* __restrict__ partial, float* __restrict__ blockSums)
{
    const int row  = blockIdx.x * 256 + threadIdx.x;   // 0..16383
    const int tile = row >> 4;
    const int r    = row & 15;

    float dp2 = 0.0f, dn2 = 0.0f;
    #pragma unroll
    for (int ks = 0; ks < KSPLIT; ++ks) {
        const float* base = partial + ((long)(tile * KSPLIT + ks) << 5);
        dp2 += base[r];
        dn2 += base[16 + r];
    }
    float x = sqrtf(dp2) - sqrtf(dn2);
    float loss = fmaxf(x, 0.0f) + log1pf(expf(-fabsf(x)));

    __shared__ float sm[256];
    sm[threadIdx.x] = loss;
    __syncthreads();
    #pragma unroll
    for (int s = 128; s > 0; s >>= 1) {
        if (threadIdx.x < s) sm[threadIdx.x] += sm[threadIdx.x + s];
        __syncthreads();
    }
    if (threadIdx.x == 0) blockSums[blockIdx.x] = sm[0];
}

// ---------------------------------------------------------------------------
// Stage 3: deterministic final reduction of the 64 block sums -> mean.
// ---------------------------------------------------------------------------
__global__ __launch_bounds__(256) void triplet_stage3(
    const float* __restrict__ blockSums, float* __restrict__ out)
{
    __shared__ float sm[256];
    const int t = threadIdx.x;
    sm[t] = (t < STAGE2_BLOCKS) ? blockSums[t] : 0.0f;
    __syncthreads();
    #pragma unroll
    for (int s = 128; s > 0; s >>= 1) {
        if (t < s) sm[t] += sm[t + s];
        __syncthreads();
    }
    if (t == 0) out[0] = sm[0] * (1.0f / (float)B_ROWS);
}

extern "C" void kernel_launch(void* const* d_in, const int* in_sizes, int n_in,
                              void* d_out, int out_size, void* d_ws, size_t ws_size,
                              hipStream_t stream) {
    const float* a = (const float*)d_in[0];
    const float* p = (const float*)d_in[1];
    const float* n = (const float*)d_in[2];
    float* out = (float*)d_out;

    float* partial   = (float*)d_ws;                    // 8192*32 floats = 1 MB
    float* blockSums = partial + (N_WAVES * 32);        // +64 floats

    const int blocks = N_WAVES / WAVES_PER_BLOCK;       // 1024 blocks x 256 thr

    triplet_stage1<<<blocks, 256, 0, stream>>>(a, p, n, partial);
    triplet_stage2<<<STAGE2_BLOCKS, 256, 0, stream>>>(partial, blockSums);
    triplet_stage3<<<1, 256, 0, stream>>>(blockSums, out);
}